// ParallelESNDriver_49323404427865
// MI455X (gfx1250) — compile-verified
//
#include <hip/hip_runtime.h>
#include <cstdint>

// Problem constants (match reference)
#define CHUNKS   16
#define RES_DIM  10000
#define NSE      2000000
#define LEAK     0.6f
#define BIAS     1.6f

// Decomposition: 16 x 50 = 800 blocks, each owns a 40,000-entry COO slice
// (40,000 * 4B = 160,000 B -> 16B-aligned; 10,000 float4 groups, no remainder).
#define BLOCKS_PER_CHUNK 50
#define SLICE   (NSE / BLOCKS_PER_CHUNK)   // 40000
#define THREADS 256                        // 8 wave32 waves per block

// True clang vector types so __builtin_nontemporal_load is legal.
typedef float vfloat4 __attribute__((ext_vector_type(4)));
typedef int   vint4   __attribute__((ext_vector_type(4)));

// ---------------------------------------------------------------------------
// Kernel 1: zero the per-chunk y accumulator in workspace (graph-safe).
// ---------------------------------------------------------------------------
__global__ void esn_zero_kernel(float* __restrict__ y, int n) {
    int i = blockIdx.x * blockDim.x + threadIdx.x;
    if (i < n) y[i] = 0.0f;
}

// ---------------------------------------------------------------------------
// Kernel 2: COO SpMV slice. x staged to LDS via gfx1250 async global->LDS
// copies (ASYNCcnt), scatter-add into a 40KB LDS accumulator with ds_add_f32,
// NT-hinted b128 streaming of the one-shot COO arrays, sparse atomic merge.
// ---------------------------------------------------------------------------
__global__ __launch_bounds__(THREADS)
void esn_spmv_kernel(const float* __restrict__ wr_vals,
                     const int*   __restrict__ wr_rows,
                     const int*   __restrict__ wr_cols,
                     const float* __restrict__ res_state,
                     float*       __restrict__ y_global) {
    __shared__ float x_lds[RES_DIM];   // 40 KB: chunk state vector
    __shared__ float y_lds[RES_DIM];   // 40 KB: block-local accumulator

    const int chunk = blockIdx.y;
    const int tid   = threadIdx.x;

    // --- Stage x into LDS with async global->LDS b128 copies (ASYNCcnt). ---
    // 10000 floats = 2500 x 16B transfers; per-lane LDS dest = low 32 bits of
    // the generic pointer (ISA 10.2: generic-LDS addr low word is LDS offset).
    const float* xg = res_state + (size_t)chunk * RES_DIM;
    for (int i = tid; i < RES_DIM / 4; i += THREADS) {
        unsigned     lds_addr = (unsigned)(uintptr_t)(&x_lds[4 * i]);
        const float* gaddr    = xg + 4 * i;
        asm volatile("global_load_async_to_lds_b128 %0, %1, off"
                     :
                     : "v"(lds_addr), "v"(gaddr)
                     : "memory");
    }
    // Zero the LDS accumulator while the async DMA is in flight.
    for (int i = tid; i < RES_DIM; i += THREADS) y_lds[i] = 0.0f;

    asm volatile("s_wait_asynccnt 0" ::: "memory");  // per-wave async complete
    __syncthreads();                                 // LDS visible block-wide

    // --- Stream this block's COO slice: NT-hinted 16B loads (one-shot data,
    //     keep L2 free for the reused x / y working set). ---
    const size_t   base = (size_t)chunk * NSE + (size_t)blockIdx.x * SLICE;
    const vfloat4* v4 = (const vfloat4*)(wr_vals + base);
    const vint4*   r4 = (const vint4*)(wr_rows + base);
    const vint4*   c4 = (const vint4*)(wr_cols + base);

    const int ngroups = SLICE / 4;  // 10000
    for (int g = tid; g < ngroups; g += THREADS) {
        vfloat4 v = __builtin_nontemporal_load(&v4[g]);
        vint4   r = __builtin_nontemporal_load(&r4[g]);
        vint4   c = __builtin_nontemporal_load(&c4[g]);
        // Banked LDS atomics: random scatter-add -> ds_add_f32 (no return)
        atomicAdd(&y_lds[r.x], v.x * x_lds[c.x]);
        atomicAdd(&y_lds[r.y], v.y * x_lds[c.y]);
        atomicAdd(&y_lds[r.z], v.z * x_lds[c.z]);
        atomicAdd(&y_lds[r.w], v.w * x_lds[c.w]);
    }
    __syncthreads();

    // --- Merge block-partial y into the L2-resident global accumulator. ---
    float* yg = y_global + (size_t)chunk * RES_DIM;
    for (int i = tid; i < RES_DIM; i += THREADS) {
        float v = y_lds[i];
        if (v != 0.0f) atomicAdd(&yg[i], v);  // skip-zero: identity adds elided
    }
}

// ---------------------------------------------------------------------------
// Kernel 3: out = LEAK * tanh(y + proj + BIAS) + (1-LEAK) * h
// ---------------------------------------------------------------------------
__global__ void esn_finalize_kernel(const float* __restrict__ y,
                                    const float* __restrict__ proj_vars,
                                    const float* __restrict__ res_state,
                                    float*       __restrict__ out,
                                    int n) {
    int i = blockIdx.x * blockDim.x + threadIdx.x;
    if (i < n) {
        float a = tanhf(y[i] + proj_vars[i] + BIAS);
        out[i] = LEAK * a + (1.0f - LEAK) * res_state[i];
    }
}

// ---------------------------------------------------------------------------
// Host launcher (graph-capture safe: only kernel launches on `stream`).
// ---------------------------------------------------------------------------
extern "C" void kernel_launch(void* const* d_in, const int* in_sizes, int n_in,
                              void* d_out, int out_size, void* d_ws, size_t ws_size,
                              hipStream_t stream) {
    const float* wr_vals   = (const float*)d_in[0];
    const int*   wr_rows   = (const int*)  d_in[1];
    const int*   wr_cols   = (const int*)  d_in[2];
    const float* proj_vars = (const float*)d_in[3];
    const float* res_state = (const float*)d_in[4];
    float*       out       = (float*)d_out;

    float* y = (float*)d_ws;  // CHUNKS*RES_DIM floats = 640 KB scratch

    const int n = CHUNKS * RES_DIM;  // 160000

    esn_zero_kernel<<<(n + 255) / 256, 256, 0, stream>>>(y, n);

    dim3 grid(BLOCKS_PER_CHUNK, CHUNKS);
    esn_spmv_kernel<<<grid, THREADS, 0, stream>>>(wr_vals, wr_rows, wr_cols,
                                                  res_state, y);

    esn_finalize_kernel<<<(n + 255) / 256, 256, 0, stream>>>(y, proj_vars,
                                                             res_state, out, n);
}